// BahdanauPointer_52510270161103
// MI455X (gfx1250) — compile-verified
//
#include <hip/hip_runtime.h>
#include <hip/hip_bf16.h>
#include <stdint.h>

// Problem sizes (fixed by the reference)
#define B_   4
#define LQ_  256
#define LK_  512
#define D_   256   // DQ == DK
#define P_   256   // PROJ
#define PPAD 260   // LDS row stride (floats): 260%64==4 -> conflict-free rows; 260*4%16==0 -> float4-aligned

typedef __attribute__((ext_vector_type(2))) float v2f;
typedef __attribute__((ext_vector_type(8))) float v8f;

__device__ __forceinline__ float fast_tanh(float x) {
#if __has_builtin(__builtin_amdgcn_tanhf)
  return __builtin_amdgcn_tanhf(x);        // v_tanh_f32 (TRANS)
#else
  // tanh(x) = 1 - 2/(exp(2x)+1): v_exp_f32 + v_rcp_f32, safe at both tails
  float e = __expf(2.0f * x);
  return fmaf(-2.0f, __builtin_amdgcn_rcpf(e + 1.0f), 1.0f);
#endif
}

// ---------------------------------------------------------------------------
// Kernel 1: C[M,N] = A[M,K] @ W[K,N] (+ bias[N]) via V_WMMA_F32_16X16X4_F32.
// One wave (32 threads) owns one 16x16 tile of C; EXEC is all-ones as required.
//
// VGPR layouts per CDNA5 ISA 7.12.2:
//   A (16x4, MxK): lanes 0-15 row=lane,  v0=K(k0+0), v1=K(k0+1)
//                  lanes16-31 row=lane-16, v0=K(k0+2), v1=K(k0+3)
//   B (4x16, KxN): VGPR j holds row (k0+j) in lanes 0-15, row (k0+j+2) in 16-31
//   C (16x16):     VGPR j -> M=j (lanes 0-15), M=j+8 (lanes 16-31), N=lane&15
// ---------------------------------------------------------------------------
__global__ __launch_bounds__(32)
void proj_wmma_kernel(const float* __restrict__ A, const float* __restrict__ W,
                      const float* __restrict__ bias, float* __restrict__ C,
                      int M, int N, int K, int ntiles_n, int has_bias) {
  const int tile = blockIdx.x;
  const int tm   = tile / ntiles_n;
  const int tn   = tile % ntiles_n;
  const int lane = threadIdx.x;
  const int half = lane >> 4;     // 0 or 1
  const int l16  = lane & 15;
  const int ka   = half * 2;      // K sub-offset selected by lane half

  // Per-lane streaming pointers (advance by 4 along K each iteration)
  const float* ap = A + (size_t)(tm * 16 + l16) * K + ka;          // A[row][k0+ka]
  const float* bp = W + (size_t)ka * N + tn * 16 + l16;            // W[k0+ka][col]
  const size_t bstep = (size_t)4 * N;

  v8f acc = {};
#pragma unroll 4
  for (int k0 = 0; k0 < K; k0 += 4) {
    v2f a, b;
    a.x = ap[0];
    a.y = ap[1];
    b.x = bp[0];
    b.y = bp[N];
    // 8 args: (neg_a, A, neg_b, B, c_mod, C, reuse_a, reuse_b)
    acc = __builtin_amdgcn_wmma_f32_16x16x4_f32(false, a, false, b,
                                                (short)0, acc, false, false);
    ap += 4;
    bp += bstep;
  }

  const float bv = has_bias ? bias[tn * 16 + l16] : 0.0f;
  float* Crow = C + (size_t)(tm * 16 + half * 8) * N + tn * 16 + l16;
#pragma unroll
  for (int j = 0; j < 8; ++j) {
    Crow[(size_t)j * N] = acc[j] + bv;
  }
}

// ---------------------------------------------------------------------------
// Kernel 2: logits[b,q,k] = sum_p tanh(qp[b,q,p] + kp[b,k,p]) * w2[p] + b2,
// masked to -inf. (b1 was already folded into qp.)
// One 256-thread block per (b, 16q x 16k) tile; qp/kp tiles staged in LDS with
// float4 (b128) transfers; inner loop is 3x ds_load_b128 + 4x (add,tanh,fma).
// ---------------------------------------------------------------------------
__global__ __launch_bounds__(256)
void score_tanh_kernel(const float* __restrict__ qp, const float* __restrict__ kp,
                       const uint8_t* __restrict__ mask,
                       const float* __restrict__ w2, const float* __restrict__ b2p,
                       float* __restrict__ out) {
  __shared__ float sq[16 * PPAD];
  __shared__ float sk[16 * PPAD];
  __shared__ float sw[P_];

  const int ktiles = LK_ / 16;   // 32
  const int qtiles = LQ_ / 16;   // 16
  const int tile = blockIdx.x;
  const int b  = tile / (ktiles * qtiles);
  const int r  = tile % (ktiles * qtiles);
  const int tq = r / ktiles;
  const int tk = r % ktiles;

  const int tid = threadIdx.x;
  const float* qbase = qp + ((size_t)b * LQ_ + tq * 16) * P_;   // 16B aligned
  const float* kbase = kp + ((size_t)b * LK_ + tk * 16) * P_;   // 16B aligned

  // Stage tiles as float4: 1024 vec4 per tile, 4 per thread, coalesced b128.
#pragma unroll
  for (int idx = tid; idx < 16 * (P_ / 4); idx += 256) {
    const int row  = idx >> 6;       // 64 float4 per row
    const int col4 = idx & 63;
    float4 vq = ((const float4*)qbase)[row * (P_ / 4) + col4];
    float4 vk = ((const float4*)kbase)[row * (P_ / 4) + col4];
    *(float4*)(sq + row * PPAD + col4 * 4) = vq;
    *(float4*)(sk + row * PPAD + col4 * 4) = vk;
  }
  sw[tid] = w2[tid];             // blockDim.x == P_ == 256
  __syncthreads();

  const int q = tid >> 4;        // output stores coalesced along k
  const int k = tid & 15;
  const float4* sq4 = (const float4*)(sq + q * PPAD);
  const float4* sk4 = (const float4*)(sk + k * PPAD);
  const float4* sw4 = (const float4*)sw;

  float acc = 0.0f;
#pragma unroll 4
  for (int p4 = 0; p4 < P_ / 4; ++p4) {
    const float4 a = sq4[p4];
    const float4 c = sk4[p4];
    const float4 w = sw4[p4];
    acc = fmaf(fast_tanh(a.x + c.x), w.x, acc);
    acc = fmaf(fast_tanh(a.y + c.y), w.y, acc);
    acc = fmaf(fast_tanh(a.z + c.z), w.z, acc);
    acc = fmaf(fast_tanh(a.w + c.w), w.w, acc);
  }

  const int qi = tq * 16 + q;
  const int ki = tk * 16 + k;
  const size_t o = ((size_t)b * LQ_ + qi) * LK_ + ki;
  float res = acc + b2p[0];
  if (mask[o]) res = -__builtin_inff();   // jnp.where(mask, -inf, logits)
  out[o] = res;
}

// ---------------------------------------------------------------------------
extern "C" void kernel_launch(void* const* d_in, const int* in_sizes, int n_in,
                              void* d_out, int out_size, void* d_ws, size_t ws_size,
                              hipStream_t stream) {
  (void)in_sizes; (void)n_in; (void)out_size; (void)ws_size;

  const float*   query = (const float*)d_in[0];   // (B,LQ,DQ) f32
  const float*   keys  = (const float*)d_in[1];   // (B,LK,DK) f32
  const uint8_t* mask  = (const uint8_t*)d_in[2]; // (B,LQ,LK) bool (1 byte/elem)
  const float*   Wq    = (const float*)d_in[3];   // (DQ,P)
  const float*   Wk    = (const float*)d_in[4];   // (DK,P)
  const float*   b1    = (const float*)d_in[5];   // (P,)
  const float*   w2    = (const float*)d_in[6];   // (P,)
  const float*   b2    = (const float*)d_in[7];   // scalar
  float* out = (float*)d_out;

  // Workspace: qp (1 MB) | kp (2 MB)
  float* qp = (float*)d_ws;
  float* kp = qp + (size_t)B_ * LQ_ * P_;

  {
    const int M = B_ * LQ_, N = P_, K = D_, nt = N / 16;
    proj_wmma_kernel<<<(M / 16) * nt, 32, 0, stream>>>(query, Wq, b1, qp,
                                                       M, N, K, nt, /*bias=*/1);
  }
  {
    const int M = B_ * LK_, N = P_, K = D_, nt = N / 16;
    proj_wmma_kernel<<<(M / 16) * nt, 32, 0, stream>>>(keys, Wk, nullptr, kp,
                                                       M, N, K, nt, /*bias=*/0);
  }
  score_tanh_kernel<<<B_ * (LQ_ / 16) * (LK_ / 16), 256, 0, stream>>>(
      qp, kp, mask, w2, b2, out);
}